// NDiagonalLayer_66048007078594
// MI455X (gfx1250) — compile-verified
//
#include <hip/hip_runtime.h>
#include <hip/hip_bf16.h>

// ---------------------------------------------------------------------------
// Banded (15-diagonal) x @ W + bias on gfx1250 via V_WMMA_F32_16X16X4_F32.
// W in 16x16 blocks is block-tridiagonal, so each 16x16 output tile is a
// K=48 GEMM: 12 chained fp32 WMMAs. Memory-bound (~256MB @ 23.3TB/s ~ 11us);
// fp32 WMMA keeps exact reference precision and is free vs the HBM floor.
// ---------------------------------------------------------------------------

#define DIM   4096
#define MROWS 8192
#define NJT   (DIM / 16)    // 256 column tiles
#define NRS   (MROWS / 16)  // 512 row strips
#define NCC   8             // column chunks per strip
#define CHUNK (NJT / NCC)   // 32 tiles per wave
#define WB_ELEMS (NJT * 3 * 4 * 64 * 2)  // 393216 floats = 1.5 MB in d_ws

typedef __attribute__((ext_vector_type(2))) float v2f;
typedef __attribute__((ext_vector_type(8))) float v8f;

// ---------------------------------------------------------------------------
// Kernel 1: pack the block-tridiagonal W into B-fragment lane layout.
// Wb[((j*3 + t)*4 + s)*64 + lane*2 + e] = W[16*(j-1+t) + k_local, 16*j + n_local]
//   k_local = 4*s + 2*(lane>=16) + e,  n_local = lane & 15
// W[k,n]: d = k-n; d==0 -> diag_w[n]; d in [-7,-1] -> upper_w[-d-1, k];
//         d in [1,7] -> lower_w[d-1, n]; else 0. Out-of-range block rows -> 0.
// ---------------------------------------------------------------------------
__global__ void __launch_bounds__(256) pack_w_kernel(
    const float* __restrict__ diag_w, const float* __restrict__ lower_w,
    const float* __restrict__ upper_w, float* __restrict__ Wb) {
  int idx = blockIdx.x * 256 + threadIdx.x;
  if (idx >= WB_ELEMS) return;

  int e    = idx & 1;
  int lane = (idx >> 1) & 31;
  int s    = (idx >> 6) & 3;
  int rem  = idx >> 8;
  int t    = rem % 3;
  int j    = rem / 3;

  int kl = 4 * s + ((lane >= 16) ? 2 : 0) + e;  // 0..15
  int nl = lane & 15;
  int bk = j - 1 + t;                            // block row of W
  int k  = 16 * bk + kl;
  int n  = 16 * j + nl;

  float v = 0.0f;
  if (bk >= 0 && bk < NJT) {
    int d = k - n;
    if (d == 0) {
      v = diag_w[n];
    } else if (d < 0 && d >= -7) {
      v = upper_w[(-d - 1) * (DIM - 1) + k];   // W[k, k+(-d)] = upper_w[-d-1, k]
    } else if (d > 0 && d <= 7) {
      v = lower_w[(d - 1) * (DIM - 1) + n];    // W[n+d, n]   = lower_w[d-1, n]
    }
  }
  Wb[idx] = v;
}

// ---------------------------------------------------------------------------
// A-fragment (16x4 fp32, ISA layout): lane<16 holds X[lane, 4s+{0,1}],
// lane>=16 holds X[lane-16, 4s+{2,3}]. Four v2f slices cover one 16x16 block.
// ---------------------------------------------------------------------------
struct AFrag { v2f s[4]; };

__device__ __forceinline__ AFrag load_a(const float* __restrict__ x,
                                        int rs, int bk, int lane) {
  AFrag a;
  if (bk < 0 || bk >= NJT) {           // wave-uniform: EXEC stays all-1s
    v2f z; z.x = 0.0f; z.y = 0.0f;
    a.s[0] = z; a.s[1] = z; a.s[2] = z; a.s[3] = z;
    return a;
  }
  const float* p = x + (size_t)(16 * rs + (lane & 15)) * DIM
                     + 16 * bk + 2 * (lane >> 4);
  a.s[0] = *(const v2f*)(p + 0);       // 8B-aligned b64 loads
  a.s[1] = *(const v2f*)(p + 4);
  a.s[2] = *(const v2f*)(p + 8);
  a.s[3] = *(const v2f*)(p + 12);
  return a;
}

// ---------------------------------------------------------------------------
// Kernel 2: block-tridiagonal GEMM. One wave32 per (row strip, column chunk).
// Sliding 3-block A window; 12 x v_wmma_f32_16x16x4_f32 per output tile.
// ---------------------------------------------------------------------------
__global__ void __launch_bounds__(256) banded_wmma_kernel(
    const float* __restrict__ x, const float* __restrict__ Wb,
    const float* __restrict__ bias, float* __restrict__ out) {
  int lane = threadIdx.x & 31;
  int wave = threadIdx.x >> 5;
  int gw   = blockIdx.x * 8 + wave;    // 0..4095
  int rs   = gw / NCC;                 // row strip 0..511
  int cc   = gw % NCC;                 // column chunk 0..7
  int j0   = cc * CHUNK;

  AFrag prev = load_a(x, rs, j0 - 1, lane);
  AFrag cur  = load_a(x, rs, j0,     lane);

  for (int j = j0; j < j0 + CHUNK; ++j) {
    AFrag next = load_a(x, rs, j + 1, lane);

    // B fragments for the 3 W blocks x 4 K-slices (coalesced b64, hits L2).
    v2f w[12];
#pragma unroll
    for (int q = 0; q < 12; ++q)
      w[q] = *(const v2f*)(Wb + ((size_t)j * 12 + q) * 64 + 2 * lane);

    // Seed accumulator with bias (C/D layout: N = lane & 15 for all rows).
    float b = bias[16 * j + (lane & 15)];
    v8f acc = {b, b, b, b, b, b, b, b};

#pragma unroll
    for (int s = 0; s < 4; ++s)
      acc = __builtin_amdgcn_wmma_f32_16x16x4_f32(
          false, prev.s[s], false, w[0 * 4 + s], (short)0, acc, false, false);
#pragma unroll
    for (int s = 0; s < 4; ++s)
      acc = __builtin_amdgcn_wmma_f32_16x16x4_f32(
          false, cur.s[s], false, w[1 * 4 + s], (short)0, acc, false, false);
#pragma unroll
    for (int s = 0; s < 4; ++s)
      acc = __builtin_amdgcn_wmma_f32_16x16x4_f32(
          false, next.s[s], false, w[2 * 4 + s], (short)0, acc, false, false);

    // Store D: lane<16 -> rows r (M=r), lane>=16 -> rows r+8; N = lane & 15.
    float* op = out + (size_t)(16 * rs + ((lane >> 4) << 3)) * DIM
                    + 16 * j + (lane & 15);
#pragma unroll
    for (int r = 0; r < 8; ++r)
      op[(size_t)r * DIM] = acc[r];

    prev = cur;
    cur  = next;
  }
}

// ---------------------------------------------------------------------------
extern "C" void kernel_launch(void* const* d_in, const int* in_sizes, int n_in,
                              void* d_out, int out_size, void* d_ws, size_t ws_size,
                              hipStream_t stream) {
  const float* x       = (const float*)d_in[0];
  const float* diag_w  = (const float*)d_in[1];
  const float* lower_w = (const float*)d_in[2];
  const float* upper_w = (const float*)d_in[3];
  const float* bias    = (const float*)d_in[4];
  float* out = (float*)d_out;
  float* Wb  = (float*)d_ws;  // 1.5 MB packed band blocks

  pack_w_kernel<<<(WB_ELEMS + 255) / 256, 256, 0, stream>>>(
      diag_w, lower_w, upper_w, Wb);

  // 512 strips * 8 chunks = 4096 waves; 8 waves/block -> 512 blocks.
  banded_wmma_kernel<<<(NRS * NCC) / 8, 256, 0, stream>>>(x, Wb, bias, out);
}